// DC3x3_33062658244791
// MI455X (gfx1250) — compile-verified
//
#include <hip/hip_runtime.h>

typedef float v2f __attribute__((ext_vector_type(2)));
typedef float v8f __attribute__((ext_vector_type(8)));

#define NN 8
#define CC 64
#define HH 128
#define WW 128
#define TAPS 9
#define HW (HH * WW)

// ws layout (floats): [0, 36864) w1p[tap][m][ci] ; [36864, 73728) w2p[tap][c][ci]
// (contiguous: phase ph slice = ws + ph*4096, ph = 0..17) ; [73728, 74304) b2p[tap][c]
#define WS_B2P 73728

// Async global->LDS 16B copy (CDNA5 GLOBAL_LOAD_ASYNC_TO_LDS_B128, ASYNCcnt).
__device__ __forceinline__ void async_g2l_b128(unsigned ldsaddr, const float* g) {
    asm volatile("global_load_async_to_lds_b128 %0, %1, off"
                 :: "v"(ldsaddr), "v"(g) : "memory");
}
__device__ __forceinline__ void wait_async0() {    // everything landed
    asm volatile("s_wait_asynccnt 0x0" ::: "memory");
}
__device__ __forceinline__ void wait_async4() {    // all but newest 4 landed
    asm volatile("s_wait_asynccnt 0x4" ::: "memory");
}

__device__ __forceinline__ int clampi(int v, int lo, int hi) {
    return v < lo ? lo : (v > hi ? hi : v);
}

// One-time repack of weights/bias into tap-sliced contiguous layout in ws.
__global__ __launch_bounds__(256) void dc3x3_repack(
    const float* __restrict__ w1, const float* __restrict__ w2,
    const float* __restrict__ b2, float* __restrict__ ws)
{
    const int i = blockIdx.x * 256 + threadIdx.x;     // 0 .. 36863
    if (i < 64 * 64 * TAPS) {
        const int tap = i >> 12, rem = i & 4095;      // rem = m*64+ci (w1), c*64+ci (w2)
        ws[i] = w1[rem * 9 + tap];                    // w1 flat: (m*64+ci)*9 + tap
        const int c = rem >> 6, ci = rem & 63;
        ws[64 * 64 * TAPS + i] = w2[((c * 9 + tap) << 6) + ci];
    }
    if (i < TAPS * 64) {
        const int tap = i >> 6, c = i & 63;
        ws[WS_B2P + i] = b2[c * 9 + tap];
    }
}

// Fused: h = conv3x3(y,w1)+b1 ; k = conv1x1(h,w2)+b2 ; out = sum_tap k * unfold(x)
// One workgroup = one image row (8 waves x 16 pixels). All GEMMs via
// V_WMMA_F32_16X16X4_F32. Weight tap-slices stream through a ping-pong LDS
// buffer via async B128 DMA-to-LDS, pipelined one phase ahead (ASYNCcnt
// completes in order => wait<=4 proves the previous slice is resident).
__global__ __launch_bounds__(256) void dc3x3_fused_wmma(
    const float* __restrict__ x,  const float* __restrict__ y,
    const float* __restrict__ b1, const float* __restrict__ wsp,
    float* __restrict__ out)
{
    __shared__ __align__(16) float lds_w[2][64 * 64];   // 32 KB ping-pong slices
    __shared__ float lds_h[8][64 * 16];                 // 32 KB h tile staging
    __shared__ float lds_b2a[TAPS][64];                 // 2.25 KB b2p copy

    const int tid  = threadIdx.x;
    const int wave = tid >> 5;
    const int lane = tid & 31;
    const int p    = lane & 15;             // pixel within wave tile (WMMA N)
    const int half = lane >> 4;             // lane-half selects K pair / M+8

    const int blk = blockIdx.x;             // n*H + h
    const int n   = blk >> 7;
    const int h   = blk & 127;
    const int w0  = wave << 4;
    const int col = w0 + p;                 // this lane's pixel column

    // LDS byte addresses of this thread's 16B staging slots (low 32 bits of
    // flat address = wave-relative LDS offset per CDNA5 aperture mapping).
    const unsigned la0 = (unsigned)(uintptr_t)(void*)&lds_w[0][tid * 4];
    const unsigned la1 = (unsigned)(uintptr_t)(void*)&lds_w[1][tid * 4];

    // Stage phase `ph` (16 KB contiguous slice) into buffer at LDS addr `la`:
    // 4 x B128 per thread; thread t covers bytes [t*16 + i*4096 ..+15].
    auto issue_phase = [&](int ph, unsigned la) {
        const float* g = wsp + ((size_t)ph << 12) + (tid << 2);
#pragma unroll
        for (int i = 0; i < 4; ++i)
            async_g2l_b128(la + (unsigned)(i * 4096), g + (i << 10));
    };

    // ---- prologue: kick off phase 0, copy b2p once --------------------------
    issue_phase(0, la0);
    for (int idx = tid; idx < TAPS * 64; idx += 256)
        (&lds_b2a[0][0])[idx] = wsp[WS_B2P + idx];

    v8f acc[4];                             // GEMM1 accumulators (h tile)
#pragma unroll
    for (int j = 0; j < 4; ++j)
#pragma unroll
        for (int r = 0; r < 8; ++r)
            acc[j][r] = b1[16 * j + 8 * half + r];

    // ================= GEMM1: h = w1 (64x576) * y-patches (576x16) + b1 ======
#pragma unroll 1
    for (int tap = 0; tap < TAPS; ++tap) {
        const int buf = tap & 1;
        __syncthreads();                    // all waves done reading buf^1
        issue_phase(tap + 1, buf ? la0 : la1);  // tap 8 prefetches phase 9 = w2 tap 0
        wait_async4();                      // in-order => this tap's slice landed
        __syncthreads();                    // slice visible to all waves

        const int dh = tap / 3 - 1, dw = tap % 3 - 1;
        const int yr = h + dh, yc = col + dw;
        const bool valid = ((unsigned)yr < (unsigned)HH) && ((unsigned)yc < (unsigned)WW);
        // clamp -> loads always in-bounds -> branchless (v_cndmask, no exec dance)
        const float* ybase = y + (size_t)n * CC * HW
                               + (size_t)clampi(yr, 0, HH - 1) * WW
                               + clampi(yc, 0, WW - 1);
        const float* lw = &lds_w[buf][0];

#pragma unroll
        for (int ks = 0; ks < 16; ++ks) {
            const int ci = ks * 4 + half * 2;     // K pair for this lane-half
            const float v0 = ybase[(size_t)ci * HW];
            const float v1 = ybase[(size_t)(ci + 1) * HW];
            v2f bf;
            bf[0] = valid ? v0 : 0.f;
            bf[1] = valid ? v1 : 0.f;
#pragma unroll
            for (int j = 0; j < 4; ++j) {
                const int mrow = 16 * j + p;      // A: M = m0 + lane%16
                v2f af;
                af[0] = lw[mrow * 64 + ci];
                af[1] = lw[mrow * 64 + ci + 1];
                acc[j] = __builtin_amdgcn_wmma_f32_16x16x4_f32(
                    false, af, false, bf, (short)0, acc[j], false, false);
            }
        }
    }

    // Transpose h tile D-layout -> B-fragment layout through per-wave LDS.
    float* hst = &lds_h[wave][0];
#pragma unroll
    for (int j = 0; j < 4; ++j)
#pragma unroll
        for (int r = 0; r < 8; ++r)
            hst[(16 * j + 8 * half + r) * 16 + p] = acc[j][r];

    v2f hfrag[16];                              // K=64 of h, kept in VGPRs
#pragma unroll
    for (int ks = 0; ks < 16; ++ks) {
        const int k = ks * 4 + half * 2;
        hfrag[ks][0] = hst[k * 16 + p];
        hfrag[ks][1] = hst[(k + 1) * 16 + p];
    }

    v8f oacc[4];                                // final output accumulators
#pragma unroll
    for (int j = 0; j < 4; ++j)
#pragma unroll
        for (int r = 0; r < 8; ++r)
            oacc[j][r] = 0.f;

    // ====== GEMM2 (w2 tap-sliced, M = channels) + dynamic-conv combine =======
    // Phase 9 (w2 tap 0) already in flight to buffer 1 (issued at GEMM1 tap 8).
#pragma unroll 1
    for (int tap = 0; tap < TAPS; ++tap) {
        const int buf = (tap + 1) & 1;      // tap0 -> buf1, tap1 -> buf0, ...
        __syncthreads();                    // all waves done reading buf^1
        if (tap + 1 < TAPS) {
            issue_phase(TAPS + 1 + tap, buf ? la0 : la1);
            wait_async4();
        } else {
            wait_async0();
        }
        __syncthreads();

        const int dh = tap / 3 - 1, dw = tap % 3 - 1;
        const int xr = h + dh, xc = col + dw;
        const bool valid = ((unsigned)xr < (unsigned)HH) && ((unsigned)xc < (unsigned)WW);
        const float* xbase = x + (size_t)n * CC * HW
                               + (size_t)clampi(xr, 0, HH - 1) * WW
                               + clampi(xc, 0, WW - 1);
        const float* lw = &lds_w[buf][0];

#pragma unroll
        for (int j = 0; j < 4; ++j) {
            v8f kf;                         // kernel frag, channels 16j..16j+15
#pragma unroll
            for (int r = 0; r < 8; ++r)
                kf[r] = lds_b2a[tap][16 * j + 8 * half + r];
#pragma unroll
            for (int ks = 0; ks < 16; ++ks) {
                const int ci = ks * 4 + half * 2;
                const int crow = 16 * j + p;
                v2f af;
                af[0] = lw[crow * 64 + ci];
                af[1] = lw[crow * 64 + ci + 1];
                kf = __builtin_amdgcn_wmma_f32_16x16x4_f32(
                    false, af, false, hfrag[ks], (short)0, kf, false, false);
            }
            // D register r holds kernel[c = 16j+8*half+r][pixel p] — same
            // layout as oacc, so the dynamic conv is a per-lane FMA.
#pragma unroll
            for (int r = 0; r < 8; ++r) {
                const int c = 16 * j + 8 * half + r;
                const float xl = xbase[(size_t)c * HW];   // clamped, in-bounds
                const float xv = valid ? xl : 0.f;        // v_cndmask
                oacc[j][r] += kf[r] * xv;
            }
        }
    }

    // Store output tile (D layout -> NCHW)
    float* obase = out + (size_t)n * CC * HW + (size_t)h * WW + col;
#pragma unroll
    for (int j = 0; j < 4; ++j)
#pragma unroll
        for (int r = 0; r < 8; ++r)
            obase[(size_t)(16 * j + 8 * half + r) * HW] = oacc[j][r];
}

extern "C" void kernel_launch(void* const* d_in, const int* in_sizes, int n_in,
                              void* d_out, int out_size, void* d_ws, size_t ws_size,
                              hipStream_t stream) {
    (void)in_sizes; (void)n_in; (void)out_size; (void)ws_size;
    const float* x  = (const float*)d_in[0];
    const float* y  = (const float*)d_in[1];
    const float* w1 = (const float*)d_in[2];
    const float* b1 = (const float*)d_in[3];
    const float* w2 = (const float*)d_in[4];
    const float* b2 = (const float*)d_in[5];
    float* out = (float*)d_out;
    float* ws  = (float*)d_ws;              // needs 74304 floats (~297 KB)

    // 1) repack weights into tap-sliced contiguous layout (one small kernel)
    dc3x3_repack<<<dim3(144), dim3(256), 0, stream>>>(w1, w2, b2, ws);
    // 2) fused WMMA kernel: one workgroup per image row (8 waves x 16 pixels)
    dc3x3_fused_wmma<<<dim3(NN * HH), dim3(256), 0, stream>>>(x, y, b1, ws, out);
}